// HoyerSparsityLoss_14104672600175
// MI455X (gfx1250) — compile-verified
//
#include <hip/hip_runtime.h>
#include <math.h>

typedef __attribute__((ext_vector_type(2))) float v2f;
typedef __attribute__((ext_vector_type(4))) float v4f;
typedef __attribute__((ext_vector_type(8))) float v8f;

#define NROWS 512
#define DIM   768
#define KC    64
#define LDST  68                 // KC + 4 pad: float4-aligned stores, bank-scattered reads
#define TEMP_INV 20.0f           // 1 / 0.05
#define EPS 1e-8f
#define SQRT_D 27.712812921102035f
#define INV_SD_M1 0.03743516082f // 1 / (sqrt(768) - 1)

// ---------------- Kernel 1: row squared-norms for all three matrices ----------------
__global__ __launch_bounds__(128)
void hoyer_norms_kernel(const float* __restrict__ A, const float* __restrict__ P,
                        const float* __restrict__ Ng, float* __restrict__ ws) {
    int wave = (blockIdx.x * blockDim.x + threadIdx.x) >> 5;   // 1536 waves total
    int lane = threadIdx.x & 31;
    int mat  = wave / NROWS;
    int row  = wave % NROWS;
    const float* src = (mat == 0) ? A : ((mat == 1) ? P : Ng);
    float s = 0.f;
    for (int k = lane; k < DIM; k += 32) {
        float v = src[row * DIM + k];
        s = fmaf(v, v, s);
    }
    #pragma unroll
    for (int m = 16; m >= 1; m >>= 1) s += __shfl_xor(s, m, 32);
    if (lane == 0) ws[mat * NROWS + row] = s;
}

// ---------------- Kernel 2: zero the denominator accumulators ----------------
__global__ __launch_bounds__(256)
void hoyer_zero_kernel(float* __restrict__ denom) {
    denom[blockIdx.x * 256 + threadIdx.x] = 0.f;
}

// ---------------- Kernel 3: fused tile kernel (WMMA dot + VALU L1 + exp + rowsum) ----------------
__global__ __launch_bounds__(32)
void hoyer_tile_kernel(const float* __restrict__ A, const float* __restrict__ P,
                       const float* __restrict__ Ng, float* __restrict__ ws) {
    __shared__ float as_[16 * LDST];
    __shared__ float bs_[16 * LDST];

    const float* nrmA  = ws;                 // [0, 512)   ||anchor||^2
    float*       denom = ws + 3 * NROWS;     // [1536,2048) sum of exp terms (pos+neg)
    float*       hdiag = ws + 4 * NROWS;     // [2048,2560) diagonal Hoyer scores (pos)

    int bx = blockIdx.x;                     // 2048 tiles: {matrix, I, J}
    int m  = bx >> 10;                       // 0 = pos, 1 = neg
    int I  = (bx >> 5) & 31;
    int J  = bx & 31;
    const float* B    = m ? Ng : P;
    const float* nrmB = ws + (m ? 2 * NROWS : NROWS);

    int lane  = threadIdx.x;
    int half  = lane >> 4;                   // lane half selects M block / K pair
    int iloc  = lane & 15;                   // A-operand row, B/C column (N)
    int rbase = half * 8;                    // C-layout: VGPR r holds M = rbase + r

    v8f   acc = {0.f, 0.f, 0.f, 0.f, 0.f, 0.f, 0.f, 0.f};
    float l1[8] = {0.f, 0.f, 0.f, 0.f, 0.f, 0.f, 0.f, 0.f};

    const float* Abase = A + (I * 16) * DIM;
    const float* Bbase = B + (J * 16) * DIM;

    for (int k0 = 0; k0 < DIM; k0 += KC) {
        // ---- cooperative stage: 16 rows x KC floats per matrix, float4 coalesced ----
        #pragma unroll
        for (int idx = lane; idx < 16 * (KC / 4); idx += 32) {
            int r  = idx >> 4;               // KC/4 == 16 float4 per row
            int c4 = idx & 15;
            v4f av = *(const v4f*)(Abase + r * DIM + k0 + c4 * 4);
            v4f bv = *(const v4f*)(Bbase + r * DIM + k0 + c4 * 4);
            *(v4f*)(as_ + r * LDST + c4 * 4) = av;
            *(v4f*)(bs_ + r * LDST + c4 * 4) = bv;
        }
        __syncthreads();

        // ---- prefetch next K chunk (global_prefetch_b8) ----
        if (k0 + KC < DIM) {
            const float* pf = (lane < 16) ? (Abase + lane * DIM + k0 + KC)
                                          : (Bbase + (lane - 16) * DIM + k0 + KC);
            __builtin_prefetch(pf, 0, 3);
        }

        // ---- WMMA f32 16x16x4 chain: acc += Atile x Btile^T ----
        // A operand: lanes 0-15 K={k,k+1}, lanes 16-31 K={k+2,k+3}, row = lane&15
        // B operand (4x16, row=K striped across lanes): same per-lane-half K split
        #pragma unroll
        for (int kk = 0; kk < KC; kk += 4) {
            int kb = kk + 2 * half;
            v2f a_op = { as_[iloc * LDST + kb], as_[iloc * LDST + kb + 1] };
            v2f b_op = { bs_[iloc * LDST + kb], bs_[iloc * LDST + kb + 1] };
            acc = __builtin_amdgcn_wmma_f32_16x16x4_f32(
                      false, a_op, false, b_op, (short)0, acc, false, false);
        }

        // ---- L1 accumulation (irreducible VALU work, co-executes with WMMA) ----
        // lane covers pairs (M = rbase+r, N = iloc); a reads broadcast per half-wave
        #pragma unroll 4
        for (int kk = 0; kk < KC; ++kk) {
            float bv = bs_[iloc * LDST + kk];
            const float* ap = as_ + rbase * LDST + kk;
            #pragma unroll
            for (int r = 0; r < 8; ++r) {
                float av = ap[r * LDST];
                l1[r] += fabsf(av - bv);
            }
        }
        __syncthreads();
    }

    // ---- epilogue: Hoyer score, exp, row-sum reduce, diagonal capture ----
    float nb_j = nrmB[J * 16 + iloc];
    #pragma unroll
    for (int r = 0; r < 8; ++r) {
        int   gi   = I * 16 + rbase + r;
        float na_i = nrmA[gi];
        float d2   = fmaxf(na_i + nb_j - 2.f * acc[r], 0.f);
        float l2   = sqrtf(d2) + EPS;
        float h    = (SQRT_D - l1[r] / l2) * INV_SD_M1;
        float e    = __expf(h * TEMP_INV);

        // sum over the 16 columns of this tile row (width-16 butterfly)
        float s = e;
        s += __shfl_xor(s, 1, 32);
        s += __shfl_xor(s, 2, 32);
        s += __shfl_xor(s, 4, 32);
        s += __shfl_xor(s, 8, 32);
        if (iloc == 0) atomicAdd(&denom[gi], s);

        if (m == 0 && I == J && iloc == rbase + r) hdiag[gi] = h;   // diagonal of matrix_pos
    }
}

// ---------------- Kernel 4: final loss = mean(-h_ii/T + log(denom_i)) ----------------
__global__ __launch_bounds__(256)
void hoyer_finalize_kernel(const float* __restrict__ ws, float* __restrict__ out) {
    __shared__ float red[256];
    const float* denom = ws + 3 * NROWS;
    const float* hdiag = ws + 4 * NROWS;
    int t = threadIdx.x;
    float s = 0.f;
    for (int i = t; i < NROWS; i += 256)
        s += __logf(denom[i]) - hdiag[i] * TEMP_INV;
    red[t] = s;
    __syncthreads();
    #pragma unroll
    for (int off = 128; off > 0; off >>= 1) {
        if (t < off) red[t] += red[t + off];
        __syncthreads();
    }
    if (t == 0) out[0] = red[0] * (1.0f / (float)NROWS);
}

extern "C" void kernel_launch(void* const* d_in, const int* in_sizes, int n_in,
                              void* d_out, int out_size, void* d_ws, size_t ws_size,
                              hipStream_t stream) {
    const float* A  = (const float*)d_in[0];
    const float* P  = (const float*)d_in[1];
    const float* Ng = (const float*)d_in[2];
    float* ws  = (float*)d_ws;      // 2560 floats: na | np | nn | denom | hdiag
    float* out = (float*)d_out;

    hoyer_norms_kernel<<<384, 128, 0, stream>>>(A, P, Ng, ws);          // 1536 waves
    hoyer_zero_kernel<<<2, 256, 0, stream>>>(ws + 3 * NROWS);           // zero 512 accumulators
    hoyer_tile_kernel<<<2048, 32, 0, stream>>>(A, P, Ng, ws);           // 2 x 32 x 32 tiles
    hoyer_finalize_kernel<<<1, 256, 0, stream>>>(ws, out);
}